// receivedEdgesToNodesAggregator_65249143161000
// MI455X (gfx1250) — compile-verified
//
#include <hip/hip_runtime.h>

// segment_sum scatter-add for MI455X (gfx1250, wave32).
//
// out[r, :] += edges[e, :] for every edge e with receivers[e] == r.
//
// Roofline: edges 320 MB (streamed once) + receivers 6.4 MB + output
// 20 MB RMW + 20 MB zero-init ~= 350 MB -> ~15 us at 23.3 TB/s. Pure
// bandwidth; WMMA cannot express a data-dependent scatter. CDNA5 angle:
// the 20 MB accumulator fits in the 192 MB L2, so all 80M f32 atomics
// are L2-resident RMWs; the edge stream uses non-temporal loads so it
// does not evict the accumulator.

#define D_EDGE 50
#define PAIRS_PER_EDGE 25  // 50 floats = 25 float2 pairs

typedef float v2f __attribute__((ext_vector_type(2)));

// ---------------------------------------------------------------------------
// Scatter-add (the hot kernel; placed first so the disasm snippet shows it).
// One thread per (edge, float2-pair): consecutive lanes load consecutive 8B
// chunks -> coalesced global_load_b64 (non-temporal). 25 consecutive lanes
// share one receivers[e] load (cache broadcast). Two native
// global_atomic_add_f32 per thread into the L2-resident output.
// ---------------------------------------------------------------------------
__global__ void rcv_scatter_add_kernel(const v2f* __restrict__ edges2,
                                       const int* __restrict__ recv,
                                       float* __restrict__ out,
                                       unsigned n_pairs) {
  unsigned idx = blockIdx.x * blockDim.x + threadIdx.x;
  if (idx >= n_pairs) return;

  unsigned e = idx / PAIRS_PER_EDGE;      // mul+shift, no real divide
  unsigned p = idx - e * PAIRS_PER_EDGE;  // pair index in row, 0..24

  // Non-temporal streaming load (TH=NT): edges are read exactly once.
  v2f v = __builtin_nontemporal_load(edges2 + idx);

  int r = recv[e];
  float* dst = out + (unsigned long long)r * D_EDGE + (unsigned long long)(p * 2u);

  // unsafeAtomicAdd -> native global_atomic_add_f32 (no CAS loop).
  unsafeAtomicAdd(dst, v.x);
  unsafeAtomicAdd(dst + 1, v.y);
}

// ---------------------------------------------------------------------------
// Zero-init, vector part: one float4 store per thread, no tail logic.
// ---------------------------------------------------------------------------
__global__ void rcv_zero4_kernel(float4* __restrict__ out4, unsigned n4) {
  unsigned i = blockIdx.x * blockDim.x + threadIdx.x;
  if (i < n4) {
    out4[i] = make_float4(0.f, 0.f, 0.f, 0.f);
  }
}

// ---------------------------------------------------------------------------
// Zero-init, scalar tail (launched only if out_size % 4 != 0; here it is 0).
// ---------------------------------------------------------------------------
__global__ void rcv_zero_tail_kernel(float* __restrict__ out, unsigned base,
                                     unsigned n) {
  unsigned i = base + threadIdx.x;
  if (i < n) out[i] = 0.f;
}

// ---------------------------------------------------------------------------
// Launch
//   d_in[0] = nodes     (float32, 100000 x 8)  -- unused (defines segments)
//   d_in[1] = edges     (float32, 1600000 x 50)
//   d_in[2] = receivers (int,     1600000)
//   d_out   = (100000 x 50) float32
// ---------------------------------------------------------------------------
extern "C" void kernel_launch(void* const* d_in, const int* in_sizes, int n_in,
                              void* d_out, int out_size, void* d_ws, size_t ws_size,
                              hipStream_t stream) {
  (void)n_in; (void)d_ws; (void)ws_size;

  const float* edges = (const float*)d_in[1];
  const int*   recv  = (const int*)d_in[2];
  float*       out   = (float*)d_out;

  const unsigned n_out   = (unsigned)out_size;                   // 5,000,000
  const unsigned n_edges = (unsigned)in_sizes[2];                // 1,600,000
  const unsigned n_pairs = n_edges * (unsigned)PAIRS_PER_EDGE;   // 40,000,000

  const unsigned threads = 256;  // 8 wave32 waves per workgroup

  // Zero the accumulator (vector body + optional scalar tail).
  {
    const unsigned n4  = n_out / 4u;
    const unsigned rem = n_out - n4 * 4u;
    if (n4) {
      const unsigned blocks = (n4 + threads - 1u) / threads;
      rcv_zero4_kernel<<<blocks, threads, 0, stream>>>((float4*)out, n4);
    }
    if (rem) {
      rcv_zero_tail_kernel<<<1, 4, 0, stream>>>(out, n4 * 4u, n_out);
    }
  }

  // Scatter-add all edges (same stream -> ordered after zeroing).
  {
    const unsigned blocks = (n_pairs + threads - 1u) / threads;
    rcv_scatter_add_kernel<<<blocks, threads, 0, stream>>>(
        (const v2f*)edges, recv, out, n_pairs);
  }
}